// StructNMaxPool2D_s_34737695490338
// MI455X (gfx1250) — compile-verified
//
#include <hip/hip_runtime.h>

typedef __attribute__((ext_vector_type(4))) float f4;

#define ROWS_TOTAL 32768     // B*C*Ho = 4*32*256
#define WAVES_PER_BLOCK 8
#define HALF_OUT 8388608     // 4*32*256*256

__device__ __forceinline__ unsigned long long uniform_base(const void* p) {
    // Wave-uniform by construction; pin into SGPRs for the GVS (saddr) form.
    unsigned long long v = (unsigned long long)p;
    unsigned lo = __builtin_amdgcn_readfirstlane((unsigned)v);
    unsigned hi = __builtin_amdgcn_readfirstlane((unsigned)(v >> 32));
    return ((unsigned long long)hi << 32) | lo;
}

// One wave32 per output row r (flat over B*C*Ho).
// Input rows 2r,2r+1 are contiguous: 4096 bytes per array per output row.
// Wave LDS region (2048 floats = 8 KiB):
//   [   0.. 511] s  row0   [ 512..1023] s  row1
//   [1024..1535] cs row0   [1536..2047] cs row1
__global__ __launch_bounds__(256) void structn_maxpool2d_kernel(
    const float* __restrict__ s, const float* __restrict__ cs,
    const float* __restrict__ wsp, const float* __restrict__ bsp,
    float* __restrict__ out_s, float* __restrict__ out_c)
{
    __shared__ __align__(16) float lds[WAVES_PER_BLOCK * 2048];

    const int wave = threadIdx.x >> 5;
    const int lane = threadIdx.x & 31;
    const int r    = blockIdx.x * WAVES_PER_BLOCK + wave;  // flat output row (wave-uniform)
    const int ch   = (r >> 8) & 31;                        // channel

    const unsigned long long sBase = uniform_base((const char*)s  + (size_t)r * 4096);
    const unsigned long long cBase = uniform_base((const char*)cs + (size_t)r * 4096);

    float* Wl = &lds[wave * 2048];
    // Low 32 bits of a flat shared pointer == LDS byte offset (ISA 10.2 aperture rule)
    const unsigned wb = (unsigned)(size_t)(void*)Wl;

    // Stage 8 KiB per wave via CDNA5 async global->LDS B128 (ASYNCcnt path),
    // GVS addressing (SGPR base + 32-bit VGPR offset), non-temporal.
#pragma unroll
    for (int k = 0; k < 8; ++k) {
        const unsigned off = (unsigned)((lane + k * 32) * 16);
        asm volatile("global_load_async_to_lds_b128 %0, %1, %2 th:TH_LOAD_NT"
                     :: "v"(wb + off),         "v"(off), "s"(sBase) : "memory");
        asm volatile("global_load_async_to_lds_b128 %0, %1, %2 th:TH_LOAD_NT"
                     :: "v"(wb + 4096u + off), "v"(off), "s"(cBase) : "memory");
    }
    asm volatile("s_wait_asynccnt 0" ::: "memory");  // wave-local data -> no barrier needed

    const float wq = wsp[ch];
    const float bq = bsp[ch];

    const f4* s0 = (const f4*)(Wl)        + lane * 4;   // 16 input cols per lane
    const f4* s1 = (const f4*)(Wl + 512)  + lane * 4;
    const f4* k0 = (const f4*)(Wl + 1024) + lane * 4;
    const f4* k1 = (const f4*)(Wl + 1536) + lane * 4;

    float rs[8], rc[8];
#pragma unroll
    for (int t = 0; t < 4; ++t) {
        const f4 a0 = s0[t], a1 = s1[t], q0 = k0[t], q1 = k1[t];
        // window 0 of this f4: input cols (.x,.y)
        {
            const float v00 = q0.x * fmaf(wq, a0.x, bq);
            const float v01 = q0.y * fmaf(wq, a0.y, bq);
            const float v10 = q1.x * fmaf(wq, a1.x, bq);
            const float v11 = q1.y * fmaf(wq, a1.y, bq);
            float best = v00, sb = a0.x, cb = q0.x;
            if (v01 > best) { best = v01; sb = a0.y; cb = q0.y; }
            if (v10 > best) { best = v10; sb = a1.x; cb = q1.x; }
            if (v11 > best) { best = v11; sb = a1.y; cb = q1.y; }
            rs[2 * t] = sb; rc[2 * t] = cb * 0.25f;
        }
        // window 1 of this f4: input cols (.z,.w)
        {
            const float v00 = q0.z * fmaf(wq, a0.z, bq);
            const float v01 = q0.w * fmaf(wq, a0.w, bq);
            const float v10 = q1.z * fmaf(wq, a1.z, bq);
            const float v11 = q1.w * fmaf(wq, a1.w, bq);
            float best = v00, sb = a0.z, cb = q0.z;
            if (v01 > best) { best = v01; sb = a0.w; cb = q0.w; }
            if (v10 > best) { best = v10; sb = a1.z; cb = q1.z; }
            if (v11 > best) { best = v11; sb = a1.w; cb = q1.w; }
            rs[2 * t + 1] = sb; rc[2 * t + 1] = cb * 0.25f;
        }
    }

    f4 oS0 = {rs[0], rs[1], rs[2], rs[3]};
    f4 oS1 = {rs[4], rs[5], rs[6], rs[7]};
    f4 oC0 = {rc[0], rc[1], rc[2], rc[3]};
    f4 oC1 = {rc[4], rc[5], rc[6], rc[7]};

    const size_t ob = (size_t)r * 256 + (size_t)lane * 8;  // 8 output cols per lane
    __builtin_nontemporal_store(oS0, (f4*)(out_s + ob));
    __builtin_nontemporal_store(oS1, (f4*)(out_s + ob) + 1);
    __builtin_nontemporal_store(oC0, (f4*)(out_c + ob));
    __builtin_nontemporal_store(oC1, (f4*)(out_c + ob) + 1);
}

extern "C" void kernel_launch(void* const* d_in, const int* in_sizes, int n_in,
                              void* d_out, int out_size, void* d_ws, size_t ws_size,
                              hipStream_t stream) {
    // setup_inputs order: d, cd, s, cs, w_s_pool, b_s_pool  (d, cd unused by reference)
    const float* s   = (const float*)d_in[2];
    const float* cs  = (const float*)d_in[3];
    const float* wsp = (const float*)d_in[4];
    const float* bsp = (const float*)d_in[5];
    float* out = (float*)d_out;  // [s_p (8388608) | cs_p/4 (8388608)]

    dim3 grid(ROWS_TOTAL / WAVES_PER_BLOCK);
    dim3 block(256);
    structn_maxpool2d_kernel<<<grid, block, 0, stream>>>(
        s, cs, wsp, bsp, out, out + HALF_OUT);
}